// RBFGIMutilAdaptiveDecoder_72748156059991
// MI455X (gfx1250) — compile-verified
//
#include <hip/hip_runtime.h>
#include <hip/hip_bf16.h>

typedef __attribute__((ext_vector_type(2))) float v2f;
typedef __attribute__((ext_vector_type(8))) float v8f;

#define NK    492      // 3 RBF kernels * 164 control points
#define NCHAN 16       // B(8) * 2 displacement channels -> WMMA N dimension
#define IMGW  512
#define IMGH  512

// ---------------------------------------------------------------------------
// Setup kernel: build alphaT[k][n] = mu + exp(0.5*var)*eps  (n = b*2 + ch)
// and cp4[k] = {cp_x, cp_y, 1/c, 0} (local control points rescaled by `scale`).
// Written into d_ws so the heavy kernel reads a precomputed table.
// ---------------------------------------------------------------------------
__global__ void rbf_setup_kernel(
    const float* __restrict__ mu0, const float* __restrict__ var0, const float* __restrict__ eps0,
    const float* __restrict__ mu1, const float* __restrict__ var1, const float* __restrict__ eps1,
    const float* __restrict__ mu2, const float* __restrict__ var2, const float* __restrict__ eps2,
    const float* __restrict__ scale_p,
    float* __restrict__ alphaT, float* __restrict__ cp4)
{
    int t = blockIdx.x * blockDim.x + threadIdx.x;

    if (t < NK * NCHAN) {
        int k = t >> 4, n = t & 15;
        int comp = k / 164, j = k - comp * 164;
        int b = n >> 1, ch = n & 1;
        const float* mu  = (comp == 0) ? mu0  : (comp == 1) ? mu1  : mu2;
        const float* var = (comp == 0) ? var0 : (comp == 1) ? var1 : var2;
        const float* eps = (comp == 0) ? eps0 : (comp == 1) ? eps1 : eps2;
        int idx = (b * 164 + j) * 2 + ch;
        alphaT[k * NCHAN + n] = mu[idx] + __expf(0.5f * var[idx]) * eps[idx];
    }

    int u = t - NK * NCHAN;
    if (u >= 0 && u < NK) {
        int comp = u / 164, j = u - comp * 164;
        float scale = scale_p[0];
        float x, y;
        if (j < 64) {                       // global 8x8 grid, 0.5 .. 7.5
            x = 0.5f + (float)(j & 7);
            y = 0.5f + (float)(j >> 3);
        } else {                            // local 10x10 grid, 1.8 .. 6.2, rescaled
            int m2 = j - 64;
            float vx = 1.8f + (float)(m2 % 10) * (4.4f / 9.0f);
            float vy = 1.8f + (float)(m2 / 10) * (4.4f / 9.0f);
            x = (vx - 4.0f) * scale + 4.0f;
            y = (vy - 4.0f) * scale + 4.0f;
        }
        float c = (comp == 0) ? 2.0f : (comp == 1) ? 1.5f : 1.0f;
        cp4[u * 4 + 0] = x;
        cp4[u * 4 + 1] = y;
        cp4[u * 4 + 2] = 1.0f / c;
        cp4[u * 4 + 3] = 0.0f;
    }
}

// ---------------------------------------------------------------------------
// Main kernel: one wave = 16 consecutive pixels, full 16 output channels.
// A (16x4 f32) = Wendland weights generated in registers, B (4x16 f32) = alpha
// slice from LDS, accumulate C (16x16 f32) across K=492 with
// v_wmma_f32_16x16x4_f32, then LDS-transpose for coalesced stores.
// ---------------------------------------------------------------------------
__global__ __launch_bounds__(256) void rbf_wmma_kernel(
    const float* __restrict__ alphaG,   // [NK][16]
    const float* __restrict__ cpG,      // [NK][4]
    float* __restrict__ out)            // [16][512][512]
{
    __shared__ float sAlpha[NK * NCHAN];    // 31.5 KB
    __shared__ float sCp[NK * 4];           //  7.9 KB
    __shared__ float sStage[NCHAN * 130];   //  8.3 KB (stride 130 -> bank spread)

    const int tid = threadIdx.x;
    for (int i = tid; i < NK * NCHAN; i += 256) sAlpha[i] = alphaG[i];
    for (int i = tid; i < NK * 4;     i += 256) sCp[i]    = cpG[i];
    __syncthreads();

    const int wave = tid >> 5;
    const int lane = tid & 31;
    const int hl   = lane >> 4;     // lane half selects which K pair this lane owns
    const int m    = lane & 15;     // A: pixel row / B: channel column

    const int h  = blockIdx.y;
    const int x0 = blockIdx.x * 128 + wave * 16;

    const float step = 8.0f / 511.0f;
    const float px = (float)(x0 + m) * step;
    const float py = (float)h * step;

    v8f acc = {};

    #pragma unroll 3
    for (int kb = 0; kb < NK; kb += 4) {
        const int k0 = kb + hl * 2;         // lanes 0-15: K=kb,kb+1; 16-31: kb+2,kb+3

        // --- generate two A-fragment weights (pure VALU, co-executes with WMMA)
        float cx0 = sCp[k0 * 4 + 0], cy0 = sCp[k0 * 4 + 1], ic0 = sCp[k0 * 4 + 2];
        float cx1 = sCp[k0 * 4 + 4], cy1 = sCp[k0 * 4 + 5], ic1 = sCp[k0 * 4 + 6];

        float dx0 = px - cx0, dy0 = py - cy0;
        float d0  = __builtin_sqrtf(dx0 * dx0 + dy0 * dy0) * ic0;
        float t0  = 1.0f - d0;
        float t0s = t0 * t0;
        float w0  = (d0 < 1.0f) ? t0s * t0s * (4.0f * d0 + 1.0f) : 0.0f;

        float dx1 = px - cx1, dy1 = py - cy1;
        float d1  = __builtin_sqrtf(dx1 * dx1 + dy1 * dy1) * ic1;
        float t1  = 1.0f - d1;
        float t1s = t1 * t1;
        float w1  = (d1 < 1.0f) ? t1s * t1s * (4.0f * d1 + 1.0f) : 0.0f;

        v2f A = { w0, w1 };
        v2f Bm = { sAlpha[k0 * NCHAN + m], sAlpha[(k0 + 1) * NCHAN + m] };

        // D = A(16x4) * B(4x16) + C, f32 throughout
        acc = __builtin_amdgcn_wmma_f32_16x16x4_f32(
            /*neg_a=*/false, A, /*neg_b=*/false, Bm,
            /*c_mod=*/(short)0, acc, /*reuse_a=*/false, /*reuse_b=*/false);
    }

    // --- stage D into LDS: sStage[channel][pixel_in_block]
    // D layout: VGPR j -> pixel rows j (lanes 0-15) and j+8 (lanes 16-31), channel = lane&15
    #pragma unroll
    for (int j = 0; j < 8; ++j) {
        int pix = wave * 16 + j + 8 * hl;
        sStage[m * 130 + pix] = acc[j];
    }
    __syncthreads();

    // --- coalesced write-out: 256 threads cover 2 planes x 128 pixels per pass
    const int p    = tid >> 7;        // 0 or 1
    const int pixl = tid & 127;
    const size_t rowbase = (size_t)h * IMGW + (size_t)blockIdx.x * 128 + pixl;
    #pragma unroll
    for (int i = 0; i < 8; ++i) {
        int plane = i * 2 + p;
        out[(size_t)plane * (IMGW * IMGH) + rowbase] = sStage[plane * 130 + pixl];
    }
}

// ---------------------------------------------------------------------------
extern "C" void kernel_launch(void* const* d_in, const int* in_sizes, int n_in,
                              void* d_out, int out_size, void* d_ws, size_t ws_size,
                              hipStream_t stream)
{
    // setup_inputs order: src, mu0, var0, eps0, mu1, var1, eps1, mu2, var2, eps2, scale
    const float* mu0  = (const float*)d_in[1];
    const float* var0 = (const float*)d_in[2];
    const float* eps0 = (const float*)d_in[3];
    const float* mu1  = (const float*)d_in[4];
    const float* var1 = (const float*)d_in[5];
    const float* eps1 = (const float*)d_in[6];
    const float* mu2  = (const float*)d_in[7];
    const float* var2 = (const float*)d_in[8];
    const float* eps2 = (const float*)d_in[9];
    const float* scale = (const float*)d_in[10];

    float* ws     = (float*)d_ws;
    float* alphaT = ws;                 // NK*16 floats
    float* cp4    = ws + NK * NCHAN;    // NK*4 floats

    int setup_threads = NK * NCHAN + NK;
    int setup_blocks  = (setup_threads + 255) / 256;
    rbf_setup_kernel<<<setup_blocks, 256, 0, stream>>>(
        mu0, var0, eps0, mu1, var1, eps1, mu2, var2, eps2, scale, alphaT, cp4);

    dim3 grid(IMGW / 128, IMGH, 1);     // 4 x 512 blocks, 8 waves each
    rbf_wmma_kernel<<<grid, 256, 0, stream>>>(alphaT, cp4, (float*)d_out);
}